// SwitchMoE_73993696576021
// MI455X (gfx1250) — compile-verified
//
#include <hip/hip_runtime.h>
#include <cstdint>

// Problem constants (match reference)
#define B_ 8
#define S_ 2048
#define D_ 1024
#define H_ 2048
#define E_ 8
#define EPS_ 1e-6f

typedef __attribute__((ext_vector_type(2))) float v2f;
typedef __attribute__((ext_vector_type(8))) float v8f;

// ---------------------------------------------------------------------------
// Kernel 0: zero the 64 MiB output (only 64 rows get overwritten later) and
// clear the 16 mask words in workspace.
// ---------------------------------------------------------------------------
__global__ void zero_out_kernel(float4* __restrict__ out4, int n4,
                                uint32_t* __restrict__ maskbits) {
  int i = blockIdx.x * blockDim.x + threadIdx.x;
  if (i < n4) out4[i] = make_float4(0.f, 0.f, 0.f, 0.f);
  if (blockIdx.x == 0 && threadIdx.x < 16) maskbits[threadIdx.x] = 0u;
}

// ---------------------------------------------------------------------------
// Kernel 1: gating. One wave per token. Lane l strides D with step 32
// (coalesced x loads), keeps 8 expert partial sums, then xor-shuffle reduce.
// Lane 0: top-2 (strict > == lax.top_k tie rule), atomicOr expert bit into
// maskbits[b][j]; for s<8 also store the full softmax row.
// ---------------------------------------------------------------------------
__global__ void __launch_bounds__(256) gate_kernel(
    const float* __restrict__ x, const float* __restrict__ Wg,
    const float* __restrict__ bg, uint32_t* __restrict__ maskbits,
    float* __restrict__ gs8) {
  const int tok = blockIdx.x * 8 + (threadIdx.x >> 5);  // 8 waves / block
  const int lane = threadIdx.x & 31;
  const float* xr = x + (size_t)tok * D_;

  float acc[8];
#pragma unroll
  for (int e = 0; e < 8; ++e) acc[e] = 0.f;

#pragma unroll 4
  for (int i = 0; i < D_ / 32; ++i) {
    const int d = i * 32 + lane;
    const float xv = xr[d];
    const float4* wr = (const float4*)(Wg + (size_t)d * E_);
    const float4 w0 = wr[0];
    const float4 w1 = wr[1];
    acc[0] += xv * w0.x; acc[1] += xv * w0.y;
    acc[2] += xv * w0.z; acc[3] += xv * w0.w;
    acc[4] += xv * w1.x; acc[5] += xv * w1.y;
    acc[6] += xv * w1.z; acc[7] += xv * w1.w;
  }

  // full-wave reduction (wave32): every lane ends with the complete sums
#pragma unroll
  for (int off = 16; off > 0; off >>= 1) {
#pragma unroll
    for (int e = 0; e < 8; ++e) acc[e] += __shfl_xor(acc[e], off, 32);
  }

  if (lane == 0) {
    float lg[8];
#pragma unroll
    for (int e = 0; e < 8; ++e) lg[e] = acc[e] + bg[e];

    // top-1
    int i0 = 0; float v0 = lg[0];
#pragma unroll
    for (int e = 1; e < 8; ++e) if (lg[e] > v0) { v0 = lg[e]; i0 = e; }
    // top-2 (skip i0; strict > keeps smallest index on ties)
    int i1 = (i0 == 0) ? 1 : 0; float v1 = lg[i1];
#pragma unroll
    for (int e = 0; e < 8; ++e)
      if (e != i0 && lg[e] > v1) { v1 = lg[e]; i1 = e; }

    const int b = tok >> 11;  // tok / S_
    atomicOr(&maskbits[b * 2 + 0], 1u << i0);
    atomicOr(&maskbits[b * 2 + 1], 1u << i1);

    const int s = tok & (S_ - 1);
    if (s < 8) {  // softmax values only needed for these tokens
      float ex[8], sum = 0.f;
#pragma unroll
      for (int e = 0; e < 8; ++e) { ex[e] = expf(lg[e] - v0); sum += ex[e]; }
      const float inv = 1.f / sum;
#pragma unroll
      for (int e = 0; e < 8; ++e) gs8[(b * 8 + s) * 8 + e] = ex[e] * inv;
    }
  }
}

// ---------------------------------------------------------------------------
// Kernel 2: tiny gate math for the 64 live (b,s) pairs.
// masked[b,s,j] = gs[b,s,j] * bit_s(maskbits[b][j]);
// denom[s,j]    = sum_b masked + EPS;  scale = 3 * sum_j masked/denom.
// ---------------------------------------------------------------------------
__global__ void scale_kernel(const float* __restrict__ gs8,
                             const uint32_t* __restrict__ maskbits,
                             float* __restrict__ scale) {
  __shared__ float m[2][8][8];  // [j][b][s]
  const int tid = threadIdx.x;  // 64 threads
  const int b = tid >> 3, s = tid & 7;
  const float g0 = gs8[(b * 8 + s) * 8 + 0];
  const float g1 = gs8[(b * 8 + s) * 8 + 1];
  const float ms0 = ((maskbits[b * 2 + 0] >> s) & 1u) ? g0 : 0.f;
  const float ms1 = ((maskbits[b * 2 + 1] >> s) & 1u) ? g1 : 0.f;
  m[0][b][s] = ms0;
  m[1][b][s] = ms1;
  __syncthreads();
  float d0 = EPS_, d1 = EPS_;
#pragma unroll
  for (int bb = 0; bb < 8; ++bb) { d0 += m[0][bb][s]; d1 += m[1][bb][s]; }
  scale[b * 8 + s] = 3.f * (ms0 / d0 + ms1 / d1);
}

// ---------------------------------------------------------------------------
// Kernel 3: h[64,2048] = gelu(x_tok @ W1^T + b1) via V_WMMA_F32_16X16X4_F32.
// One wave = one 16x16 tile. fp32 A/B layout (ISA 7.12.2): lanes0-15 carry
// K={0,1}, lanes16-31 K={2,3}; A row = token m, B row = W1 row n.
// grid = 64 blocks (4 M-tiles x 16), 8 waves/block -> 128 N-tiles.
// ---------------------------------------------------------------------------
__global__ void __launch_bounds__(256) ffn1_kernel(
    const float* __restrict__ x, const float* __restrict__ W1,
    const float* __restrict__ b1, float* __restrict__ h) {
  const int lane = threadIdx.x & 31;
  const int wid = threadIdx.x >> 5;
  const int mtile = blockIdx.x >> 4;               // 0..3
  const int ntile = (blockIdx.x & 15) * 8 + wid;   // 0..127
  const int laneM = lane & 15;
  const int laneHi = lane >> 4;
  const int m = mtile * 16 + laneM;                // token 0..63
  // token m -> (b = m>>3, s = m&7) row of x
  const float* arow = x + (size_t)((m >> 3) * S_ + (m & 7)) * D_ + 2 * laneHi;
  const float* brow = W1 + (size_t)(ntile * 16 + laneM) * D_ + 2 * laneHi;

  v8f c = {};
#pragma unroll 8
  for (int k = 0; k < D_; k += 4) {
    v2f a = *(const v2f*)(arow + k);
    v2f bmat = *(const v2f*)(brow + k);
    c = __builtin_amdgcn_wmma_f32_16x16x4_f32(false, a, false, bmat,
                                              (short)0, c, false, false);
  }

  const int col = ntile * 16 + laneM;
  const float bias = b1[col];
  const int mbase = mtile * 16 + laneHi * 8;
#pragma unroll
  for (int r = 0; r < 8; ++r) {
    float v = c[r] + bias;
    v = 0.5f * v * (1.f + erff(v * 0.70710678118654752f));  // exact GELU
    h[(size_t)(mbase + r) * H_ + col] = v;
  }
}

// ---------------------------------------------------------------------------
// Kernel 4: y = h @ W2^T + b2; out[row] = scale[row] * y (scattered to the
// 64 nonzero output rows). grid = 32 blocks (4 M-tiles x 8), 8 waves/block.
// ---------------------------------------------------------------------------
__global__ void __launch_bounds__(256) ffn2_kernel(
    const float* __restrict__ h, const float* __restrict__ W2,
    const float* __restrict__ b2, const float* __restrict__ scale,
    float* __restrict__ out) {
  const int lane = threadIdx.x & 31;
  const int wid = threadIdx.x >> 5;
  const int mtile = blockIdx.x >> 3;              // 0..3
  const int ntile = (blockIdx.x & 7) * 8 + wid;   // 0..63
  const int laneM = lane & 15;
  const int laneHi = lane >> 4;
  const int m = mtile * 16 + laneM;
  const float* arow = h + (size_t)m * H_ + 2 * laneHi;
  const float* brow = W2 + (size_t)(ntile * 16 + laneM) * H_ + 2 * laneHi;

  v8f c = {};
#pragma unroll 8
  for (int k = 0; k < H_; k += 4) {
    v2f a = *(const v2f*)(arow + k);
    v2f bmat = *(const v2f*)(brow + k);
    c = __builtin_amdgcn_wmma_f32_16x16x4_f32(false, a, false, bmat,
                                              (short)0, c, false, false);
  }

  const int col = ntile * 16 + laneM;
  const float bias = b2[col];
  const int mbase = mtile * 16 + laneHi * 8;
#pragma unroll
  for (int r = 0; r < 8; ++r) {
    const int row = mbase + r;
    const float v = (c[r] + bias) * scale[row];
    out[(size_t)((row >> 3) * S_ + (row & 7)) * D_ + col] = v;
  }
}

// ---------------------------------------------------------------------------
extern "C" void kernel_launch(void* const* d_in, const int* in_sizes, int n_in,
                              void* d_out, int out_size, void* d_ws,
                              size_t ws_size, hipStream_t stream) {
  (void)in_sizes; (void)n_in; (void)out_size; (void)ws_size;
  const float* x  = (const float*)d_in[0];
  const float* Wg = (const float*)d_in[1];
  const float* bg = (const float*)d_in[2];
  const float* W1 = (const float*)d_in[3];
  const float* b1 = (const float*)d_in[4];
  const float* W2 = (const float*)d_in[5];
  const float* b2 = (const float*)d_in[6];
  float* out = (float*)d_out;

  // workspace layout
  uint32_t* maskbits = (uint32_t*)d_ws;                     // 16 words
  float* gs8   = (float*)d_ws + 16;                         // [8][8][8]
  float* scale = (float*)d_ws + 16 + 512;                   // [64]
  float* h     = (float*)((char*)d_ws + 4096);              // [64][2048]

  const int n4 = (B_ * S_ * D_) / 4;
  zero_out_kernel<<<(n4 + 255) / 256, 256, 0, stream>>>((float4*)out, n4,
                                                        maskbits);
  gate_kernel<<<(B_ * S_) / 8, 256, 0, stream>>>(x, Wg, bg, maskbits, gs8);
  scale_kernel<<<1, 64, 0, stream>>>(gs8, maskbits, scale);
  ffn1_kernel<<<64, 256, 0, stream>>>(x, W1, b1, h);
  ffn2_kernel<<<32, 256, 0, stream>>>(h, W2, b2, scale, out);
}